// Hadamard_Transform_66838281061170
// MI455X (gfx1250) — compile-verified
//
#include <hip/hip_runtime.h>

// Hadamard transform over channel dim, C = 1024, x: (32,1024,32,32) f32.
// H1024 = (H32 kron H32)/32 ; H32[i,j] = (-1)^popc(i&j).
// Two-stage Kronecker factorization -> memory-bound (~256 MiB @ 23.3 TB/s ~= 11 us).
// Stages computed with V_WMMA_F32_16X16X4_F32 (fp32-exact, 16x16 tiles, K=4 chunks).

typedef __attribute__((ext_vector_type(2))) float v2f;
typedef __attribute__((ext_vector_type(8))) float v8f;

#define LDS_STRIDE 40              // 160B rows: 16B-aligned for b128, banks +16 offset
#define LDS_WORDS  (1024 * LDS_STRIDE)
#define LDS_BYTES  (LDS_WORDS * 4)

#if __has_builtin(__builtin_amdgcn_wmma_f32_16x16x4_f32)
#define HAVE_WMMA_F32X4 1
#endif

// +-scl via parity-of-popcount injected into the sign bit (3 VALU, no cndmask)
__device__ __forceinline__ float hsign(int i, int j, float scl) {
    unsigned p = (unsigned)__popc(i & j) << 31;   // shift keeps only parity bit
    return __uint_as_float(__float_as_uint(scl) ^ p);
}

// Both 16-wide column tiles of D[M, 0..31] = sum_{k<32} (scl*H32[m0+M,k]) * B[k, 0..31],
// B rows in LDS at base + k*kstride. A-operand shared across the two N-tiles.
// A layout (16x4 f32): lanes0-15/16-31 both M=0..15; VGPR0=K0|K2, VGPR1=K1|K3.
// B layout (4x16 f32) analogous: VGPR0 = rows K0|K2, VGPR1 = rows K1|K3.
__device__ __forceinline__ void h32_mm_row(const float* base, int kstride, int m0,
                                           int lane, float scl, v8f& c0, v8f& c1) {
    const int nlo = lane & 15;
    c0 = {};
    c1 = {};
#if defined(HAVE_WMMA_F32X4)
    const int krow = (lane >> 4) << 1;   // 0 for lanes 0-15, 2 for lanes 16-31
    #pragma unroll
    for (int kc = 0; kc < 8; ++kc) {
        const int kb = kc * 4;
        v2f a, b0, b1;
        a.x  = hsign(m0 + nlo, kb + krow,     scl);
        a.y  = hsign(m0 + nlo, kb + krow + 1, scl);
        const float* r0 = base + (kb + krow) * kstride + nlo;
        const float* r1 = r0 + kstride;
        b0.x = r0[0];   b0.y = r1[0];
        b1.x = r0[16];  b1.y = r1[16];
        // 8 args: (neg_a, A, neg_b, B, c_mod, C, reuse_a, reuse_b)
        c0 = __builtin_amdgcn_wmma_f32_16x16x4_f32(false, a, false, b0,
                                                   (short)0, c0, false, false);
        c1 = __builtin_amdgcn_wmma_f32_16x16x4_f32(false, a, false, b1,
                                                   (short)0, c1, false, false);
    }
#else
    // VALU fallback (identical data flow / D-register layout)
    const int rowhalf = (lane >> 4) * 8;
    #pragma unroll
    for (int k = 0; k < 32; ++k) {
        const float x0 = base[k * kstride + nlo];
        const float x1 = base[k * kstride + 16 + nlo];
        #pragma unroll
        for (int r = 0; r < 8; ++r) {
            const float s = hsign(m0 + r + rowhalf, k, scl);
            c0[r] += s * x0;
            c1[r] += s * x1;
        }
    }
#endif
}

__global__ __launch_bounds__(1024)
void hadamard1024_kernel(const float* __restrict__ x, float* __restrict__ out) {
    extern __shared__ float tile[];               // [1024][LDS_STRIDE]
    const int b    = blockIdx.x >> 5;             // 32 batches
    const int h    = blockIdx.x & 31;             // 32 rows
    const int lane = threadIdx.x & 31;
    const int g    = threadIdx.x >> 5;            // wave id 0..31
    const int nlo     = lane & 15;
    const int rowhalf = (lane >> 4) * 8;          // D layout: VGPR r -> M = r + rowhalf

    // ---- Phase 0: wave g loads channels [g*32, g*32+32) for row h into LDS ----
    // float4 per lane: lanes 0-7 cover one full 128B w-row; 4 rows per iteration.
    {
        const int wsub = (lane & 7) * 4;          // w start (16B chunk)
        const int rsub = lane >> 3;               // 0..3 row within group of 4
        const float* srcb = x + ((size_t)(b * 1024 + g * 32) * 1024) + (size_t)h * 32;
        #pragma unroll
        for (int i = 0; i < 8; ++i) {
            const int row = i * 4 + rsub;
            const float4 v = *(const float4*)(srcb + (size_t)row * 1024 + wsub);
            *(float4*)&tile[(g * 32 + row) * LDS_STRIDE + wsub] = v;
        }
    }

    // ---- Stage A: per wave, Y[g*32 + d0][w] = sum_{c0} H32[d0,c0] * X[g*32+c0][w] ----
    // Wave g reads/writes only its own 32 rows -> no barrier needed, in-place.
    const float* baseA = tile + g * 32 * LDS_STRIDE;
    v8f acc[2][2];
    #pragma unroll
    for (int mt = 0; mt < 2; ++mt)
        h32_mm_row(baseA, LDS_STRIDE, mt * 16, lane, 1.0f, acc[mt][0], acc[mt][1]);

    #pragma unroll
    for (int mt = 0; mt < 2; ++mt)
        #pragma unroll
        for (int nt = 0; nt < 2; ++nt)
            #pragma unroll
            for (int r = 0; r < 8; ++r)
                tile[(g * 32 + mt * 16 + r + rowhalf) * LDS_STRIDE + nt * 16 + nlo]
                    = acc[mt][nt][r];

    __syncthreads();

    // ---- Stage B: wave g (= d0): out[d1*32+g] = (1/32) sum_{c1} H32[c1,d1] * Y[c1*32+g] ----
    const float* baseB = tile + g * LDS_STRIDE;   // rows at stride 32*LDS_STRIDE
    float* dst = out + ((size_t)b * 1024) * 1024 + (size_t)h * 32;
    #pragma unroll
    for (int mt = 0; mt < 2; ++mt) {
        v8f c0, c1;
        h32_mm_row(baseB, 32 * LDS_STRIDE, mt * 16, lane, 0.03125f, c0, c1);
        #pragma unroll
        for (int r = 0; r < 8; ++r) {
            const int d1 = mt * 16 + r + rowhalf;
            const int ch = d1 * 32 + g;
            dst[(size_t)ch * 1024 + nlo]      = c0[r];
            dst[(size_t)ch * 1024 + 16 + nlo] = c1[r];
        }
    }
}

extern "C" void kernel_launch(void* const* d_in, const int* in_sizes, int n_in,
                              void* d_out, int out_size, void* d_ws, size_t ws_size,
                              hipStream_t stream) {
    (void)in_sizes; (void)n_in; (void)d_ws; (void)ws_size; (void)out_size;
    const float* x = (const float*)d_in[0];
    // d_in[1] (hadamard_matrix) unused: signs are synthesized exactly via popcount,
    // and the 1/sqrt(1024) = 1/32 scale is exact in fp32.
    float* out = (float*)d_out;
    dim3 grid(32 * 32);   // (b, h)
    dim3 block(1024);     // 32 waves: wave = c1-group (stage A) / d0 (stage B)
    hipLaunchKernelGGL(hadamard1024_kernel, grid, block, LDS_BYTES, stream, x, out);
}